// mixer_class_singleres_64037962383759
// MI455X (gfx1250) — compile-verified
//
#include <hip/hip_runtime.h>
#include <hip/hip_bf16.h>

// ---------------------------------------------------------------------------
// PointNet patch embedding for MI455X (gfx1250, wave32).
//
// Roofline reasoning:
//  * FPS: 1024 serial steps -> latency bound. Keep all 8192 points + the
//    running-min distance array in LDS (136 KB of the 320 KB WGP LDS),
//    one workgroup per batch.
//  * KNN: selection-bound; 1 thread per query, unrolled 32-entry insertion
//    list kept in VGPRs, candidates streamed from LDS.
//  * Embedding GEMM: [262144 x 6] x [6 x 128]. Only ~0.4 GFLOP but 134 MB of
//    fp32 output -> ~6 us floor at 23.3 TB/s HBM. Matrix op done with
//    V_WMMA_F32_16X16X4_F32 (fp32, K padded 6->8, two WMMAs per tile),
//    fused bias+BN+ReLU epilogue in registers, LDS-staged coalesced
//    float4 stores along the contiguous S dimension.
// ---------------------------------------------------------------------------

typedef __attribute__((ext_vector_type(2))) float v2f;
typedef __attribute__((ext_vector_type(8))) float v8f;

namespace cfg {
constexpr int B = 8;
constexpr int N = 8192;
constexpr int P = 1024;   // n_patches
constexpr int S = 32;     // n_samples
constexpr int D = 128;    // embedding dim
}

// ---------------------------------------------------------------------------
// Kernel 1: farthest point sampling. One block per batch, 1024 threads.
// Dynamic LDS: sx,sy,sz,dist (4*N floats) + reduce val (1024) + reduce idx.
// ---------------------------------------------------------------------------
__global__ __launch_bounds__(1024) void fps_kernel(
    const float* __restrict__ x,   // [B,3,N]
    int* __restrict__ fps_idx,     // [B,P]
    float* __restrict__ centers)   // [B,P,3]
{
  using namespace cfg;
  extern __shared__ float sm_fps[];
  float* sx   = sm_fps;
  float* sy   = sm_fps + N;
  float* sz   = sm_fps + 2 * N;
  float* dist = sm_fps + 3 * N;
  float* rv   = sm_fps + 4 * N;          // 1024 floats
  int*   ri   = (int*)(rv + 1024);       // 1024 ints

  const int b   = blockIdx.x;
  const int tid = threadIdx.x;

  for (int j = tid; j < N; j += 1024) {
    sx[j] = x[(b * 3 + 0) * N + j];
    sy[j] = x[(b * 3 + 1) * N + j];
    sz[j] = x[(b * 3 + 2) * N + j];
    dist[j] = 1e10f;
  }
  __syncthreads();

  int far = 0;
  for (int i = 0; i < P; ++i) {
    const float cx = sx[far], cy = sy[far], cz = sz[far];
    if (tid == 0) {
      fps_idx[b * P + i] = far;
      centers[(b * P + i) * 3 + 0] = cx;
      centers[(b * P + i) * 3 + 1] = cy;
      centers[(b * P + i) * 3 + 2] = cz;
    }
    float bestv = -1.0f;
    int   besti = 0;
    #pragma unroll
    for (int j = tid; j < N; j += 1024) {
      const float dx = sx[j] - cx, dy = sy[j] - cy, dz = sz[j] - cz;
      const float d  = dx * dx + dy * dy + dz * dz;
      const float dm = fminf(dist[j], d);
      dist[j] = dm;
      if (dm > bestv || (dm == bestv && j < besti)) { bestv = dm; besti = j; }
    }
    rv[tid] = bestv;
    ri[tid] = besti;
    __syncthreads();
    for (int off = 512; off > 0; off >>= 1) {
      if (tid < off) {
        const float ov = rv[tid + off];
        const int   oi = ri[tid + off];
        if (ov > rv[tid] || (ov == rv[tid] && oi < ri[tid])) {
          rv[tid] = ov;
          ri[tid] = oi;
        }
      }
      __syncthreads();
    }
    far = ri[0];
    __syncthreads();
  }
}

// ---------------------------------------------------------------------------
// Kernel 2: 32-NN for each center. One thread per query, xyz in LDS.
// Grid: B * (P/256) blocks of 256. Dynamic LDS: sx,sy,sz,sq (4*N floats).
// Insertion list fully unrolled -> stays in VGPRs (no scratch).
// ---------------------------------------------------------------------------
__global__ __launch_bounds__(256) void knn_kernel(
    const float* __restrict__ x,        // [B,3,N]
    const float* __restrict__ centers,  // [B,P,3]
    int* __restrict__ knn)              // [B,P,S]
{
  using namespace cfg;
  extern __shared__ float sm_knn[];
  float* sx = sm_knn;
  float* sy = sm_knn + N;
  float* sz = sm_knn + 2 * N;
  float* sq = sm_knn + 3 * N;

  const int b     = blockIdx.x / (P / 256);
  const int chunk = blockIdx.x % (P / 256);

  for (int j = threadIdx.x; j < N; j += 256) {
    const float px = x[(b * 3 + 0) * N + j];
    const float py = x[(b * 3 + 1) * N + j];
    const float pz = x[(b * 3 + 2) * N + j];
    sx[j] = px; sy[j] = py; sz[j] = pz;
    sq[j] = px * px + py * py + pz * pz;
  }
  __syncthreads();

  const int p  = chunk * 256 + threadIdx.x;
  const float qx = centers[(b * P + p) * 3 + 0];
  const float qy = centers[(b * P + p) * 3 + 1];
  const float qz = centers[(b * P + p) * 3 + 2];
  const float qsq = qx * qx + qy * qy + qz * qz;

  float bd[S];
  int   bi[S];
  #pragma unroll
  for (int k = 0; k < S; ++k) { bd[k] = 3.4e38f; bi[k] = 0; }

  #pragma unroll 1
  for (int j = 0; j < N; ++j) {
    const float d = qsq + sq[j] - 2.0f * (qx * sx[j] + qy * sy[j] + qz * sz[j]);
    if (d < bd[S - 1]) {
      bd[S - 1] = d;
      bi[S - 1] = j;
      #pragma unroll
      for (int k = S - 1; k > 0; --k) {
        if (bd[k] < bd[k - 1]) {
          const float tv = bd[k]; bd[k] = bd[k - 1]; bd[k - 1] = tv;
          const int   ti = bi[k]; bi[k] = bi[k - 1]; bi[k - 1] = ti;
        }
      }
    }
  }
  #pragma unroll
  for (int k = 0; k < S; ++k) knn[(b * P + p) * S + k] = bi[k];
}

// ---------------------------------------------------------------------------
// Kernel 3: gather neighbors, build 6-dim features, WMMA fp32 GEMM to 128
// channels, fused bias + BN(eval) + ReLU, LDS transpose, coalesced stores.
// One wave handles a 16(M-rows) x 128(N-channels) output tile via 8 C
// fragments and 2 V_WMMA_F32_16X16X4_F32 per fragment (K = 6 padded to 8).
// Since 16 | S, all 16 rows of a tile share (b, p) -> shared center, and the
// 16 output floats per channel are contiguous in the [B,D,P,S] output.
// ---------------------------------------------------------------------------
__global__ __launch_bounds__(256) void embed_kernel(
    const float* __restrict__ x,        // [B,3,N]
    const float* __restrict__ W,        // [D,6]
    const float* __restrict__ bias,     // [D]
    const float* __restrict__ gamma,    // [D]
    const float* __restrict__ beta,     // [D]
    const float* __restrict__ rmean,    // [D]
    const float* __restrict__ rvar,     // [D]
    const float* __restrict__ centers,  // [B,P,3]
    const int* __restrict__ knn,        // [B,P,S]
    float* __restrict__ out)            // [B,D,P,S]
{
  using namespace cfg;
  // 8 waves/block, each wave stages its 128x16 tile: 8 * 2048 * 4B = 64 KB.
  __shared__ __align__(16) float tile[8][D * 16];

  const int wave = threadIdx.x >> 5;
  const int lane = threadIdx.x & 31;
  const bool lo  = lane < 16;
  const int ln16 = lane & 15;

  const int mtile = blockIdx.x * 8 + wave;         // 16384 tiles total
  const int r0    = mtile * 16;                    // first flat row (b,p,s)
  const int b_    = r0 / (P * S);
  const int p_    = (r0 / S) % P;
  const int s0    = r0 % S;                        // 0 or 16

  // --- B fragments for all 8 channel tiles (layout: V0 = K0|K2, V1 = K1|K3)
  v2f bf0[8], bf1[8];
  #pragma unroll
  for (int t = 0; t < 8; ++t) {
    const float* wrow = W + (t * 16 + ln16) * 6;
    if (lo) {
      bf0[t].x = wrow[0]; bf0[t].y = wrow[1];      // K = 0,1
      bf1[t].x = wrow[4]; bf1[t].y = wrow[5];      // K = 4,5
    } else {
      bf0[t].x = wrow[2]; bf0[t].y = wrow[3];      // K = 2,3
      bf1[t].x = 0.0f;    bf1[t].y = 0.0f;         // K = 6,7 (pad)
    }
  }

  // --- A fragment: feature row for lane's M (both half-waves share M=ln16)
  const float cx = centers[(b_ * P + p_) * 3 + 0];
  const float cy = centers[(b_ * P + p_) * 3 + 1];
  const float cz = centers[(b_ * P + p_) * 3 + 2];
  const int   nb = knn[(b_ * P + p_) * S + s0 + ln16];
  const float nx = x[(b_ * 3 + 0) * N + nb];
  const float ny = x[(b_ * 3 + 1) * N + nb];
  const float nz = x[(b_ * 3 + 2) * N + nb];
  // feat = [rel.x, rel.y, rel.z, ctr.x, ctr.y, ctr.z]
  v2f a0, a1;
  if (lo) {
    a0.x = nx - cx; a0.y = ny - cy;                // K = 0,1
    a1.x = cy;      a1.y = cz;                     // K = 4,5
  } else {
    a0.x = nz - cz; a0.y = cx;                     // K = 2,3
    a1.x = 0.0f;    a1.y = 0.0f;                   // K = 6,7 (pad)
  }

  // --- WMMA accumulate: D = A(16x8) x B(8x128) as 8 tiles of 2 K-steps
  v8f acc[8];
  #pragma unroll
  for (int t = 0; t < 8; ++t) {
    v8f c = {};
    c = __builtin_amdgcn_wmma_f32_16x16x4_f32(false, a0, false, bf0[t],
                                              (short)0, c, false, false);
    c = __builtin_amdgcn_wmma_f32_16x16x4_f32(false, a1, false, bf1[t],
                                              (short)0, c, false, false);
    acc[t] = c;
  }

  // --- fused bias + BN(eval) + ReLU epilogue, stage into LDS as [n][m]
  #pragma unroll
  for (int t = 0; t < 8; ++t) {
    const int   n  = t * 16 + ln16;
    const float sc = gamma[n] * rsqrtf(rvar[n] + 1e-5f);
    const float sh = beta[n] - rmean[n] * sc;
    const float bn = bias[n];
    #pragma unroll
    for (int r = 0; r < 8; ++r) {
      const int m = r + (lo ? 0 : 8);
      float v = (acc[t][r] + bn) * sc + sh;
      v = fmaxf(v, 0.0f);
      tile[wave][n * 16 + m] = v;
    }
  }
  __syncthreads();

  // --- coalesced float4 stores: per channel n, 16 contiguous floats at
  //     out[((b*D + n)*P + p)*S + s0 .. +15]
  const float4* src = (const float4*)tile[wave];
  #pragma unroll
  for (int i = lane; i < D * 4; i += 32) {         // 512 float4 per wave
    const int n = i >> 2;
    const int q = i & 3;
    const size_t dst = ((size_t)(b_ * D + n) * P + p_) * S + s0 + q * 4;
    *(float4*)(out + dst) = src[i];
  }
}

// ---------------------------------------------------------------------------
// Host-side launcher.
// ---------------------------------------------------------------------------
extern "C" void kernel_launch(void* const* d_in, const int* in_sizes, int n_in,
                              void* d_out, int out_size, void* d_ws, size_t ws_size,
                              hipStream_t stream) {
  using namespace cfg;
  const float* x     = (const float*)d_in[0];
  const float* W     = (const float*)d_in[1];
  const float* bias  = (const float*)d_in[2];
  const float* gamma = (const float*)d_in[3];
  const float* beta  = (const float*)d_in[4];
  const float* rmean = (const float*)d_in[5];
  const float* rvar  = (const float*)d_in[6];
  float* out = (float*)d_out;

  // workspace layout
  char* ws = (char*)d_ws;
  int*   fps_idx = (int*)ws;                                  // B*P ints
  float* centers = (float*)(ws + sizeof(int) * B * P);        // B*P*3 floats
  int*   knn     = (int*)(ws + sizeof(int) * B * P
                             + sizeof(float) * B * P * 3);    // B*P*S ints

  // FPS: one block per batch; LDS = (4*N + 1024)*4 + 1024*4 bytes = 136 KB
  const size_t fps_lds = (size_t)(4 * N + 1024) * sizeof(float)
                       + (size_t)1024 * sizeof(int);
  fps_kernel<<<B, 1024, fps_lds, stream>>>(x, fps_idx, centers);

  // KNN: B * (P/256) blocks of 256; LDS = 4*N*4 = 128 KB
  const size_t knn_lds = (size_t)(4 * N) * sizeof(float);
  knn_kernel<<<B * (P / 256), 256, knn_lds, stream>>>(x, centers, knn);

  // Embedding: 16384 M-tiles / 8 waves per block -> 2048 blocks of 256
  embed_kernel<<<2048, 256, 0, stream>>>(x, W, bias, gamma, beta,
                                         rmean, rvar, centers, knn, out);
}